// PSRoiAlignPooling_42167988912554
// MI455X (gfx1250) — compile-verified
//
#include <hip/hip_runtime.h>
#include <stdint.h>

#define POOLK 7
#define ALPHA 32
#define Himg  128
#define Wimg  128
#define CHN   (POOLK * POOLK * ALPHA)   // 1568
#define NBIN  (POOLK * POOLK)           // 49
#define DEPTH 3                         // TDM pipeline depth (buffers)

#if defined(__AMDGCN__) && __has_builtin(__builtin_amdgcn_tensor_load_to_lds) && __has_builtin(__builtin_amdgcn_s_wait_tensorcnt)
#define USE_TDM 1
#else
#define USE_TDM 0
#endif

typedef __attribute__((ext_vector_type(4))) uint32_t u32x4;
typedef __attribute__((ext_vector_type(8))) int32_t  i32x8;
typedef __attribute__((ext_vector_type(4))) int32_t  i32x4;

// One interpolation axis: floor, frac weight, clipped neighbor indices, validity.
__device__ __forceinline__ void axis_interp(float s, int n, int& i0, int& i1,
                                            float& w, bool& v) {
  float f = floorf(s);
  w = s - f;
  int i = (int)f;
  i0 = i < 0 ? 0 : (i > n - 1 ? n - 1 : i);
  i1 = (i0 + 1 > n - 1) ? n - 1 : i0 + 1;
  v = (s >= 0.0f) && (s <= (float)(n - 1));
}

__global__ __launch_bounds__(32) void psroialign_kernel(
    const float* __restrict__ img, const float* __restrict__ rois,
    float* __restrict__ out) {
  const int roi  = blockIdx.x;   // 0..nroi-1
  const int bx   = blockIdx.y;   // x-bin 0..6  (g = bx*7 + by)
  const int lane = threadIdx.x;  // channel within group (alpha), 0..31

  const float xmin = rois[4 * roi + 0];
  const float ymin = rois[4 * roi + 1];
  const float xmax = rois[4 * roi + 2];
  const float ymax = rois[4 * roi + 3];
  const float step_x = (xmax - xmin) * (1.0f / POOLK);
  const float step_y = (ymax - ymin) * (1.0f / POOLK);

  // x side is fixed for this block (two sample columns of the bin)
  const float x1c = xmin + (float)bx * step_x;
  const float sx0 = x1c * (float)(Wimg - 1);
  const float sx1 = (x1c + step_x) * (float)(Wimg - 1);
  int ix0[2], ix1[2];
  float wx[2];
  bool vx[2];
  axis_interp(sx0, Wimg, ix0[0], ix1[0], wx[0], vx[0]);
  axis_interp(sx1, Wimg, ix0[1], ix1[1], wx[1], vx[1]);

  // 16 gathered pixel indices for one bin: s = ty*2+tx, corners (cy,cx)
  auto make_pix = [&](int by, int* pix) {
    float y1c = ymin + (float)by * step_y;
    float sy0 = y1c * (float)(Himg - 1);
    float sy1 = (y1c + step_y) * (float)(Himg - 1);
    int iy0[2], iy1[2];
    float wy[2];
    bool vy[2];
    axis_interp(sy0, Himg, iy0[0], iy1[0], wy[0], vy[0]);
    axis_interp(sy1, Himg, iy0[1], iy1[1], wy[1], vy[1]);
#pragma unroll
    for (int ty = 0; ty < 2; ++ty)
#pragma unroll
      for (int tx = 0; tx < 2; ++tx) {
        int s = ty * 2 + tx;
        pix[s * 4 + 0] = iy0[ty] * Wimg + ix0[tx];
        pix[s * 4 + 1] = iy0[ty] * Wimg + ix1[tx];
        pix[s * 4 + 2] = iy1[ty] * Wimg + ix0[tx];
        pix[s * 4 + 3] = iy1[ty] * Wimg + ix1[tx];
      }
  };

  // Bilinear combine + validity mask + 4-sample max from 16 fetched values.
  auto combine = [&](int by, const float f[16]) -> float {
    float y1c = ymin + (float)by * step_y;
    float sy0 = y1c * (float)(Himg - 1);
    float sy1 = (y1c + step_y) * (float)(Himg - 1);
    int iy0[2], iy1[2];
    float wy[2];
    bool vy[2];
    axis_interp(sy0, Himg, iy0[0], iy1[0], wy[0], vy[0]);
    axis_interp(sy1, Himg, iy0[1], iy1[1], wy[1], vy[1]);
    float best = 0.0f;
#pragma unroll
    for (int ty = 0; ty < 2; ++ty)
#pragma unroll
      for (int tx = 0; tx < 2; ++tx) {
        int s = ty * 2 + tx;
        float Wy = wy[ty], Wx = wx[tx];
        float v = f[s * 4 + 0] * ((1.0f - Wy) * (1.0f - Wx));
        v = fmaf(f[s * 4 + 1], (1.0f - Wy) * Wx, v);
        v = fmaf(f[s * 4 + 2], Wy * (1.0f - Wx), v);
        v = fmaf(f[s * 4 + 3], Wy * Wx, v);
        v = (vy[ty] && vx[tx]) ? v : 0.0f;
        best = (s == 0) ? v : fmaxf(best, v);
      }
    return best;
  };

#if USE_TDM
  // TDM gather-mode staging: 16 rows x 32 fp32 (128B) per bin, triple buffered.
  // 6KB LDS/block * 32 WG/WGP = 192KB < 320KB -> occupancy unchanged.
  __shared__ float buf[DEPTH][16][ALPHA];

  auto tdm_issue = [&](int by, float* ldsdst) {
    int pix[16];
    make_pix(by, pix);
    uint32_t ldsoff = (uint32_t)(uintptr_t)ldsdst;  // low 32 bits = LDS offset
    uint64_t ga = (uint64_t)(uintptr_t)(img + (size_t)(bx * POOLK + by) * ALPHA);
    // D# group0: count=1, gather_mode=1, 16-bit indices; lds addr; 57b global; type=2
    u32x4 d0 = {0x80000001u, ldsoff, (uint32_t)ga,
                ((uint32_t)(ga >> 32) & 0x01FFFFFFu) | 0x80000000u};
    // D# group1: data_size=4B(code 2); tensor_dim0=CHN; tensor_dim1=H*W;
    // tile_dim0=32 elements; tile_dim1=16 valid indices; dim0 stride=CHN.
    i32x8 d1 = {(int)(2u << 16),
                (int)((uint32_t)(CHN & 0xffff) << 16),
                (int)(((uint32_t)((Himg * Wimg) & 0xffff) << 16) | (CHN >> 16)),
                (int)(((uint32_t)32 << 16) | ((Himg * Wimg) >> 16)),
                16,
                CHN,
                0,
                0};
    int p[16];
#pragma unroll
    for (int i = 0; i < 16; ++i) p[i] = __builtin_amdgcn_readfirstlane(pix[i]);
    i32x4 d2 = {p[0] | (p[1] << 16), p[2] | (p[3] << 16),
                p[4] | (p[5] << 16), p[6] | (p[7] << 16)};
    i32x4 d3 = {p[8] | (p[9] << 16), p[10] | (p[11] << 16),
                p[12] | (p[13] << 16), p[14] | (p[15] << 16)};
    i32x8 dz = {0, 0, 0, 0, 0, 0, 0, 0};  // 6-arg toolchain: extra group, unused
    __builtin_amdgcn_tensor_load_to_lds(d0, d1, d2, d3, dz, 0);
  };

  // Prologue: fill the pipeline (bins 0..DEPTH-1 in flight).
  tdm_issue(0, &buf[0][0][0]);
  tdm_issue(1, &buf[1][0][0]);
  tdm_issue(2, &buf[2][0][0]);

#pragma unroll
  for (int by = 0; by < POOLK; ++by) {
    // In-order TDM completion: ops outstanding beyond bin 'by' = min(2, 6-by).
    if (by < POOLK - 2)      __builtin_amdgcn_s_wait_tensorcnt(2);
    else if (by == POOLK - 2) __builtin_amdgcn_s_wait_tensorcnt(1);
    else                      __builtin_amdgcn_s_wait_tensorcnt(0);

    const float* vb = &buf[by % DEPTH][0][0];
    float f[16];
#pragma unroll
    for (int i = 0; i < 16; ++i) f[i] = vb[i * ALPHA + lane];  // conflict-free
    float r = combine(by, f);
    size_t oidx = ((size_t)roi * NBIN + (size_t)(bx * POOLK + by)) * ALPHA + lane;
    __builtin_nontemporal_store(r, out + oidx);  // keep image hot in L2

    // Refill: bin by's LDS values already consumed -> safe to reuse its buffer.
    if (by + DEPTH < POOLK) tdm_issue(by + DEPTH, &buf[(by + DEPTH) % DEPTH][0][0]);
  }
#else
  // Fallback: direct coalesced gathers (128B per request across the wave).
  for (int by = 0; by < POOLK; ++by) {
    int pix[16];
    make_pix(by, pix);
    const int g = bx * POOLK + by;
    const size_t cbase = (size_t)g * ALPHA + lane;
    float f[16];
#pragma unroll
    for (int i = 0; i < 16; ++i) f[i] = img[(size_t)pix[i] * CHN + cbase];
    float r = combine(by, f);
    size_t oidx = ((size_t)roi * NBIN + (size_t)g) * ALPHA + lane;
    __builtin_nontemporal_store(r, out + oidx);
  }
#endif
}

extern "C" void kernel_launch(void* const* d_in, const int* in_sizes, int n_in,
                              void* d_out, int out_size, void* d_ws, size_t ws_size,
                              hipStream_t stream) {
  const float* img  = (const float*)d_in[0];
  const float* rois = (const float*)d_in[1];
  float* out = (float*)d_out;
  (void)d_ws; (void)ws_size; (void)out_size; (void)n_in;
  const int nroi = in_sizes[1] / 4;   // rois: (1, N, 4)
  dim3 grid((unsigned)nroi, POOLK);
  psroialign_kernel<<<grid, 32, 0, stream>>>(img, rois, out);
}